// MeshReduce_89472758710376
// MI455X (gfx1250) — compile-verified
//
#include <hip/hip_runtime.h>
#include <hip/hip_bf16.h>

// ---- problem constants (match reference) ----
#define BB  32
#define NM  131072
#define PP  256
#define FF  3
#define EE  64
#define HH  4
#define DH  16

typedef float v2f __attribute__((ext_vector_type(2)));
typedef float v8f __attribute__((ext_vector_type(8)));

// -------------------------------------------------------------------------
// Kernel 1: KNN pivotal <- mesh. One block per pivotal point (256 blocks).
// Each thread scans 512 mesh positions, keeps a sorted top-3, then LDS tree
// merge. Emits indices + inverse-squared-distance weights (clip 1e-16).
// -------------------------------------------------------------------------
__global__ void knn_pivotal_kernel(const float* __restrict__ posM,
                                   const float* __restrict__ posP,
                                   int* __restrict__ idx1,
                                   float* __restrict__ w1) {
    __shared__ float rd[256][3];
    __shared__ int   ri[256][3];
    const int p = blockIdx.x;
    const int t = threadIdx.x;
    const float px = posP[p * 2 + 0];
    const float py = posP[p * 2 + 1];

    float d0 = 3.4e38f, d1 = 3.4e38f, d2 = 3.4e38f;
    int   i0 = 0,       i1 = 0,       i2 = 0;
    for (int j = t; j < NM; j += 256) {
        __builtin_prefetch(posM + 2 * (j + 4096), 0, 0);   // global_prefetch_b8
        const float dx = posM[2 * j + 0] - px;
        const float dy = posM[2 * j + 1] - py;
        const float d  = dx * dx + dy * dy;
        if (d < d2) {
            if (d < d0)      { d2 = d1; i2 = i1; d1 = d0; i1 = i0; d0 = d; i0 = j; }
            else if (d < d1) { d2 = d1; i2 = i1; d1 = d;  i1 = j; }
            else             { d2 = d;  i2 = j; }
        }
    }
    rd[t][0] = d0; rd[t][1] = d1; rd[t][2] = d2;
    ri[t][0] = i0; ri[t][1] = i1; ri[t][2] = i2;
    __syncthreads();

#define KNN_INS(dv, iv)                                                     \
    if ((dv) < c2) {                                                        \
        if ((dv) < c0)      { c2 = c1; z2 = z1; c1 = c0; z1 = z0; c0 = (dv); z0 = (iv); } \
        else if ((dv) < c1) { c2 = c1; z2 = z1; c1 = (dv); z1 = (iv); }     \
        else                { c2 = (dv); z2 = (iv); }                       \
    }

    for (int off = 128; off > 0; off >>= 1) {
        if (t < off) {
            float c0 = rd[t][0], c1 = rd[t][1], c2 = rd[t][2];
            int   z0 = ri[t][0], z1 = ri[t][1], z2 = ri[t][2];
            const float b0 = rd[t + off][0], b1 = rd[t + off][1], b2 = rd[t + off][2];
            const int   x0 = ri[t + off][0], x1 = ri[t + off][1], x2 = ri[t + off][2];
            KNN_INS(b0, x0); KNN_INS(b1, x1); KNN_INS(b2, x2);
            rd[t][0] = c0; rd[t][1] = c1; rd[t][2] = c2;
            ri[t][0] = z0; ri[t][1] = z1; ri[t][2] = z2;
        }
        __syncthreads();
    }
    if (t == 0) {
        for (int k = 0; k < 3; ++k) {
            idx1[p * 3 + k] = ri[0][k];
            w1[p * 3 + k]   = 1.0f / fmaxf(rd[0][k], 1e-16f);
        }
    }
}

// -------------------------------------------------------------------------
// Kernel 2: fused LayerNorm(F=3) + inverse-d^2 interpolation -> y [B,P,3].
// LN is only evaluated at the 768 gathered mesh rows per batch (the full-
// tensor LN in the reference is dead work).
// -------------------------------------------------------------------------
__global__ void gather_ln_interp_kernel(const float* __restrict__ na,
                                        const int* __restrict__ idx1,
                                        const float* __restrict__ w1,
                                        const float* __restrict__ gamma,
                                        const float* __restrict__ beta,
                                        float* __restrict__ y) {
    const int tid = blockIdx.x * blockDim.x + threadIdx.x;
    if (tid >= BB * PP) return;
    const int b = tid >> 8;
    const int p = tid & 255;
    const float g0 = gamma[0], g1 = gamma[1], g2 = gamma[2];
    const float e0b = beta[0], e1b = beta[1], e2b = beta[2];
    float n0 = 0.f, n1 = 0.f, n2 = 0.f, den = 0.f;
    for (int k = 0; k < 3; ++k) {
        const int   i = idx1[p * 3 + k];
        const float w = w1[p * 3 + k];
        const float* a = na + ((long long)b * NM + i) * 3;
        const float a0 = a[0], a1 = a[1], a2 = a[2];
        const float mu = (a0 + a1 + a2) * (1.0f / 3.0f);
        const float c0 = a0 - mu, c1 = a1 - mu, c2 = a2 - mu;
        const float var = (c0 * c0 + c1 * c1 + c2 * c2) * (1.0f / 3.0f);
        const float rs  = rsqrtf(var + 1e-5f);
        n0 += w * (c0 * rs * g0 + e0b);
        n1 += w * (c1 * rs * g1 + e1b);
        n2 += w * (c2 * rs * g2 + e2b);
        den += w;
    }
    const float rden = 1.0f / den;
    y[tid * 3 + 0] = n0 * rden;
    y[tid * 3 + 1] = n1 * rden;
    y[tid * 3 + 2] = n2 * rden;
}

// -------------------------------------------------------------------------
// Kernel 3: q[bp,e] = y[bp,:]@Wf[e,:] + bf[e] + posP[p,:]@Wp[e,:] + bp[e].
// K=3 / K=2 contractions -> scalar FMA, one thread per output element.
// -------------------------------------------------------------------------
__global__ void q_embed_kernel(const float* __restrict__ y,
                               const float* __restrict__ posP,
                               const float* __restrict__ Wf,
                               const float* __restrict__ bf,
                               const float* __restrict__ Wp,
                               const float* __restrict__ bpv,
                               float* __restrict__ q) {
    const int gid = blockIdx.x * blockDim.x + threadIdx.x;
    if (gid >= BB * PP * EE) return;
    const int bp = gid >> 6;
    const int e  = gid & 63;
    const int p  = bp & 255;
    float v = bf[e] + bpv[e];
    v += y[bp * 3 + 0] * Wf[e * 3 + 0];
    v += y[bp * 3 + 1] * Wf[e * 3 + 1];
    v += y[bp * 3 + 2] * Wf[e * 3 + 2];
    v += posP[p * 2 + 0] * Wp[e * 2 + 0];
    v += posP[p * 2 + 1] * Wp[e * 2 + 1];
    q[gid] = v;
}

// -------------------------------------------------------------------------
// Kernel 4: Q = q@Wq^T + bq, K = q@Wk^T + bk via V_WMMA_F32_16X16X4_F32.
// M = B*P = 8192, N = 64, Kdim = 64 (16 k-steps of 4). One wave per 16x16
// (Mtile,Ntile); A fragment shared between the Q and K accumulators.
// A layout: lane m=l&15, kh=l>>4 -> a = {A[m][2kh], A[m][2kh+1]} per k-step.
// D layout: lane holds D[v + 8*kh][l&15], v=0..7.
// -------------------------------------------------------------------------
__global__ void qk_proj_wmma_kernel(const float* __restrict__ q,
                                    const float* __restrict__ inW,
                                    const float* __restrict__ inB,
                                    float* __restrict__ Qm,
                                    float* __restrict__ Km) {
    const int wid  = (blockIdx.x * blockDim.x + threadIdx.x) >> 5;  // 0..2047
    const int lane = threadIdx.x & 31;
    const int mt = wid >> 2;            // 0..511
    const int nt = wid & 3;             // 0..3
    const int M0 = mt * 16, N0 = nt * 16;
    const int m  = lane & 15;
    const int kh = lane >> 4;

    const float* qrow  = q   + (M0 + m) * EE;
    const float* wqrow = inW + (N0 + m) * EE;          // Wq row n (n == m per lane)
    const float* wkrow = inW + (EE + N0 + m) * EE;     // Wk row n

    v8f accQ = {};
    v8f accK = {};
#pragma unroll
    for (int ks = 0; ks < 16; ++ks) {
        const int e0 = ks * 4 + kh * 2;
        v2f a  = { qrow[e0],  qrow[e0 + 1]  };
        v2f bq = { wqrow[e0], wqrow[e0 + 1] };
        v2f bk = { wkrow[e0], wkrow[e0 + 1] };
        accQ = __builtin_amdgcn_wmma_f32_16x16x4_f32(false, a, false, bq, (short)0, accQ, false, false);
        accK = __builtin_amdgcn_wmma_f32_16x16x4_f32(false, a, false, bk, (short)0, accK, false, false);
    }

    const float biasQ = inB[N0 + m];
    const float biasK = inB[EE + N0 + m];
#pragma unroll
    for (int v = 0; v < 8; ++v) {
        const int row = M0 + v + 8 * kh;
        Qm[row * EE + N0 + m] = accQ[v] + biasQ;
        Km[row * EE + N0 + m] = accK[v] + biasK;
    }
}

// -------------------------------------------------------------------------
// Kernel 5: attention. Grid = B * (P/16) blocks of 128 threads (4 waves =
// 4 heads). scores[h] (16x256) computed with WMMA into LDS, scaled by
// 1/sqrt(DH); per-wave softmax with wave32 shfl reductions (1/H head-mean
// folded into the normalizer); then fused attn_w @ y -> y2 [B,P,3].
// -------------------------------------------------------------------------
__global__ void attn_kernel(const float* __restrict__ Qm,
                            const float* __restrict__ Km,
                            const float* __restrict__ y,
                            float* __restrict__ y2) {
    __shared__ float s[HH][16][PP];   // 64 KB of the WGP's 320 KB LDS
    const int b  = blockIdx.x >> 4;
    const int q0 = (blockIdx.x & 15) * 16;
    const int h    = threadIdx.x >> 5;    // wave == head
    const int lane = threadIdx.x & 31;
    const int m  = lane & 15;
    const int kh = lane >> 4;

    const float* qbase = Qm + ((b * PP + q0 + m) * EE + h * DH);
    for (int kt = 0; kt < 16; ++kt) {
        const float* kbase = Km + ((b * PP + kt * 16 + m) * EE + h * DH);
        v8f acc = {};
#pragma unroll
        for (int ks = 0; ks < 4; ++ks) {
            const int d0 = ks * 4 + kh * 2;
            v2f a  = { qbase[d0], qbase[d0 + 1] };
            v2f bb = { kbase[d0], kbase[d0 + 1] };
            acc = __builtin_amdgcn_wmma_f32_16x16x4_f32(false, a, false, bb, (short)0, acc, false, false);
        }
#pragma unroll
        for (int v = 0; v < 8; ++v)
            s[h][v + 8 * kh][kt * 16 + m] = acc[v] * 0.25f;   // 1/sqrt(DH)
    }
    __syncthreads();

    // softmax along k for this head's 16 rows; fold 1/H into normalizer
    for (int r = 0; r < 16; ++r) {
        float vals[8];
        float mx = -3.4e38f;
#pragma unroll
        for (int i = 0; i < 8; ++i) {
            vals[i] = s[h][r][lane + 32 * i];
            mx = fmaxf(mx, vals[i]);
        }
        for (int off = 16; off; off >>= 1) mx = fmaxf(mx, __shfl_xor(mx, off, 32));
        float sum = 0.f;
#pragma unroll
        for (int i = 0; i < 8; ++i) { vals[i] = __expf(vals[i] - mx); sum += vals[i]; }
        for (int off = 16; off; off >>= 1) sum += __shfl_xor(sum, off, 32);
        const float rn = 0.25f / sum;                         // 1/H head mean
#pragma unroll
        for (int i = 0; i < 8; ++i) s[h][r][lane + 32 * i] = vals[i] * rn;
    }
    __syncthreads();

    // y2[b, q0+r, f] = sum_k (sum_h s[h][r][k]) * y[b, k, f]
    if (threadIdx.x < 48) {
        const int r = threadIdx.x / 3;
        const int f = threadIdx.x % 3;
        const float* yb = y + b * PP * 3 + f;
        float acc = 0.f;
        for (int k = 0; k < PP; ++k) {
            const float aw = s[0][r][k] + s[1][r][k] + s[2][r][k] + s[3][r][k];
            acc += aw * yb[k * 3];
        }
        y2[(b * PP + q0 + r) * 3 + f] = acc;
    }
}

// -------------------------------------------------------------------------
// Kernel 6: KNN mesh <- pivotal + scatter of interpolated y2 to all batches.
// Pivotal positions cached in LDS; y2 (96 KB) stays hot in L2. Streams the
// 50 MB output -- this is the bandwidth-bound tail of the whole pipeline.
// -------------------------------------------------------------------------
__global__ void knn_mesh_out_kernel(const float* __restrict__ posM,
                                    const float* __restrict__ posP,
                                    const float* __restrict__ y2,
                                    float* __restrict__ out) {
    __shared__ float ppx[PP], ppy[PP];
    const int t = threadIdx.x;
    ppx[t] = posP[t * 2 + 0];
    ppy[t] = posP[t * 2 + 1];
    __syncthreads();

    const int j = blockIdx.x * 256 + t;
    const float mx = posM[j * 2 + 0];
    const float my = posM[j * 2 + 1];

    float c0 = 3.4e38f, c1 = 3.4e38f, c2 = 3.4e38f;
    int   z0 = 0,       z1 = 0,       z2 = 0;
    for (int p = 0; p < PP; ++p) {
        const float dx = ppx[p] - mx;
        const float dy = ppy[p] - my;
        const float d  = dx * dx + dy * dy;
        KNN_INS(d, p);
    }
    float w0 = 1.0f / fmaxf(c0, 1e-16f);
    float w1 = 1.0f / fmaxf(c1, 1e-16f);
    float w2 = 1.0f / fmaxf(c2, 1e-16f);
    const float rden = 1.0f / (w0 + w1 + w2);
    w0 *= rden; w1 *= rden; w2 *= rden;

    for (int b = 0; b < BB; ++b) {
        const float* yb = y2 + b * PP * 3;
        float* op = out + ((long long)b * NM + j) * 3;
        op[0] = w0 * yb[z0 * 3 + 0] + w1 * yb[z1 * 3 + 0] + w2 * yb[z2 * 3 + 0];
        op[1] = w0 * yb[z0 * 3 + 1] + w1 * yb[z1 * 3 + 1] + w2 * yb[z2 * 3 + 1];
        op[2] = w0 * yb[z0 * 3 + 2] + w1 * yb[z1 * 3 + 2] + w2 * yb[z2 * 3 + 2];
    }
}
#undef KNN_INS

// -------------------------------------------------------------------------
extern "C" void kernel_launch(void* const* d_in, const int* in_sizes, int n_in,
                              void* d_out, int out_size, void* d_ws, size_t ws_size,
                              hipStream_t stream) {
    const float* node_attr = (const float*)d_in[0];
    const float* posM      = (const float*)d_in[1];
    const float* posP      = (const float*)d_in[2];
    const float* ln_gamma  = (const float*)d_in[3];
    const float* ln_beta   = (const float*)d_in[4];
    const float* Wf        = (const float*)d_in[5];
    const float* bf        = (const float*)d_in[6];
    const float* Wp        = (const float*)d_in[7];
    const float* bpv       = (const float*)d_in[8];
    const float* in_proj_w = (const float*)d_in[9];
    const float* in_proj_b = (const float*)d_in[10];
    float* out = (float*)d_out;

    // workspace carve-up (256B-aligned offsets, ~6.5 MB total)
    char* ws = (char*)d_ws;
    int*   idx1 = (int*)  (ws + 0);                         //   3 KB
    float* w1   = (float*)(ws + 4096);                      //   3 KB
    float* y    = (float*)(ws + 8192);                      //  96 KB  [B,P,3]
    float* q    = (float*)(ws + 8192 + 98304);              //   2 MB  [B*P,64]
    float* Qm   = (float*)(ws + 106496 + 2097152);          //   2 MB
    float* Km   = (float*)(ws + 2203648 + 2097152);         //   2 MB
    float* y2   = (float*)(ws + 4300800 + 2097152);         //  96 KB  [B,P,3]

    knn_pivotal_kernel<<<PP, 256, 0, stream>>>(posM, posP, idx1, w1);
    gather_ln_interp_kernel<<<(BB * PP + 255) / 256, 256, 0, stream>>>(
        node_attr, idx1, w1, ln_gamma, ln_beta, y);
    q_embed_kernel<<<(BB * PP * EE + 255) / 256, 256, 0, stream>>>(
        y, posP, Wf, bf, Wp, bpv, q);
    // 2048 (Mtile,Ntile) waves / 8 waves per 256-thread block
    qk_proj_wmma_kernel<<<256, 256, 0, stream>>>(q, in_proj_w, in_proj_b, Qm, Km);
    attn_kernel<<<BB * (PP / 16), 128, 0, stream>>>(Qm, Km, y, y2);
    knn_mesh_out_kernel<<<NM / 256, 256, 0, stream>>>(posM, posP, y2, out);
}